// SparseAdapter_73160472920581
// MI455X (gfx1250) — compile-verified
//
#include <hip/hip_runtime.h>
#include <hip/hip_bf16.h>

#define Hd 2048
#define BN 64
#define TOPK 128
#define Bb 4
#define Tt 4096

typedef __attribute__((ext_vector_type(2))) float v2f;
typedef __attribute__((ext_vector_type(4))) float v4f;
typedef __attribute__((ext_vector_type(8))) float v8f;

// ---------------------------------------------------------------------------
// Kernel 1: fused  out = x  (stream copy)  +  logits = x . Wg + bg
// One wave (32 lanes) per token row of 2048 floats (16 x float4 per lane).
// Bulk streams are non-temporal (TH=NT) so the 8KB Wg stays hot in cache.
// ---------------------------------------------------------------------------
__global__ __launch_bounds__(256) void copy_logits_kernel(
    const float* __restrict__ x, const float* __restrict__ Wg,
    const float* __restrict__ bg, float* __restrict__ out,
    float* __restrict__ logits) {
  const int wave = (blockIdx.x * blockDim.x + threadIdx.x) >> 5;  // token id
  const int lane = threadIdx.x & 31;
  const v4f* xr  = (const v4f*)(x + (size_t)wave * Hd);
  v4f*       orw = (v4f*)(out + (size_t)wave * Hd);
  const v4f* wg4 = (const v4f*)Wg;
  float acc = 0.f;
#pragma unroll
  for (int i = 0; i < Hd / 128; ++i) {
    v4f a = __builtin_nontemporal_load(&xr[i * 32 + lane]);
    v4f w = wg4[i * 32 + lane];
    __builtin_nontemporal_store(a, &orw[i * 32 + lane]);
    acc += a.x * w.x + a.y * w.y + a.z * w.z + a.w * w.w;
  }
#pragma unroll
  for (int off = 16; off > 0; off >>= 1) acc += __shfl_xor(acc, off, 32);
  if (lane == 0) logits[wave] = acc + bg[0];
}

// ---------------------------------------------------------------------------
// Kernel 2: per-batch top-128 of 4096 logits via 4-pass radix select on the
// monotonic (sign-flipped) uint ordering. Strictly-greater values are emitted
// unordered (the adapter only needs the set); exact-threshold ties are taken
// by smallest index, matching jax.lax.top_k.
// ---------------------------------------------------------------------------
__global__ __launch_bounds__(1024) void topk_kernel(
    const float* __restrict__ logits, int* __restrict__ selected) {
  __shared__ unsigned vals[Tt];
  __shared__ int hist[256];
  __shared__ unsigned sprefix;
  __shared__ int sremaining, gcnt, eqcnt;
  __shared__ int eqidx[Tt];

  const int b = blockIdx.x;
  const float* lg = logits + b * Tt;
  for (int i = threadIdx.x; i < Tt; i += 1024) {
    unsigned u = __float_as_uint(lg[i]);
    vals[i] = (u & 0x80000000u) ? ~u : (u | 0x80000000u);  // ascending-orderable
  }
  if (threadIdx.x == 0) {
    sprefix = 0; sremaining = TOPK; gcnt = 0; eqcnt = 0;
  }
  __syncthreads();

  for (int pass = 0; pass < 4; ++pass) {
    const int shift = 24 - 8 * pass;
    for (int i = threadIdx.x; i < 256; i += 1024) hist[i] = 0;
    __syncthreads();
    const unsigned pref = sprefix;  // pass*8 high bits already decided
    for (int i = threadIdx.x; i < Tt; i += 1024) {
      unsigned v = vals[i];
      if (pass == 0 || (v >> (shift + 8)) == pref)
        atomicAdd(&hist[(v >> shift) & 0xFFu], 1);
    }
    __syncthreads();
    if (threadIdx.x == 0) {
      int rem = sremaining, acc = 0, bin = 255;
      for (; bin > 0; --bin) {            // descending cumulative count
        if (acc + hist[bin] >= rem) break;
        acc += hist[bin];
      }
      sremaining = rem - acc;             // take this many from chosen bin
      sprefix = (pref << 8) | (unsigned)bin;
    }
    __syncthreads();
  }

  const unsigned thr = sprefix;  // exact 32-bit k-th largest (orderable)
  const int rem = sremaining;    // # of ==thr entries to keep (smallest idx)
  for (int i = threadIdx.x; i < Tt; i += 1024) {
    unsigned v = vals[i];
    if (v > thr) {
      int p = atomicAdd(&gcnt, 1);        // p < TOPK - rem by construction
      selected[b * TOPK + p] = b * Tt + i;
    } else if (v == thr) {
      int p = atomicAdd(&eqcnt, 1);
      eqidx[p] = i;
    }
  }
  __syncthreads();
  if (threadIdx.x == 0) {
    // typical case: eqcnt == rem == 1; partial selection for pathological ties
    int base = gcnt, n = eqcnt;
    for (int s = 0; s < rem; ++s) {
      int mi = s;
      for (int t2 = s + 1; t2 < n; ++t2)
        if (eqidx[t2] < eqidx[mi]) mi = t2;
      int tmp = eqidx[s]; eqidx[s] = eqidx[mi]; eqidx[mi] = tmp;
      selected[b * TOPK + base + s] = b * Tt + eqidx[s];
    }
  }
}

// ---------------------------------------------------------------------------
// Kernel 3: sparse adapter on the 512 selected tokens, 16-token tiles.
//   Hs = relu(Xs[16,2048] @ Wd[2048,64] + bd)   (wave w owns N-cols w*16..+15)
//   out[tile rows] += Hs[16,64] @ Wu[64,2048] + bu
// Both GEMMs via V_WMMA_F32_16X16X4_F32 (fp32 matrix core path).
// A-frag (16x4 f32): lanes 0-15 M=0..15 K={0,1}; lanes 16-31 M=0..15 K={2,3}.
// C/D: VGPR r, lanes 0-15 -> M=r, N=lane; lanes 16-31 -> M=r+8, N=lane-16.
// ---------------------------------------------------------------------------
__global__ __launch_bounds__(128) void adapter_kernel(
    const float* __restrict__ x, const float* __restrict__ Wd,
    const float* __restrict__ bd, const float* __restrict__ Wu,
    const float* __restrict__ bu, const int* __restrict__ selected,
    float* __restrict__ out) {
  __shared__ int gtok[16];
  __shared__ float hs[16 * 65];  // stride 65: conflict-free column reads

  const int tile = blockIdx.x;        // 0..31  (32 tiles x 16 tokens)
  const int w    = threadIdx.x >> 5;  // wave 0..3
  const int lane = threadIdx.x & 31;
  const int half = lane >> 4;         // 0: K+{0,1}, 1: K+{2,3}
  const int l16  = lane & 15;
  const int koffA = half * 2;

  if (threadIdx.x < 16) gtok[threadIdx.x] = selected[tile * 16 + threadIdx.x];
  __syncthreads();

  const float* xrow = x + (size_t)gtok[l16] * Hd;  // A-row for this lane

  // ---- GEMM1: c = Xs @ Wd  (K = 2048, 512 wmma steps) ----
  v8f c = {};
  for (int k0 = 0; k0 < Hd; k0 += 4) {
    v2f a;
    a.x = xrow[k0 + koffA];
    a.y = xrow[k0 + koffA + 1];
    v2f bf;  // B[k][n] = Wd[k*BN + w*16 + n]
    bf.x = Wd[(k0 + koffA) * BN + w * 16 + l16];
    bf.y = Wd[(k0 + koffA + 1) * BN + w * 16 + l16];
    c = __builtin_amdgcn_wmma_f32_16x16x4_f32(false, a, false, bf,
                                              (short)0, c, false, false);
  }
  // bias + relu -> LDS
  const float bdv = bd[w * 16 + l16];
#pragma unroll
  for (int r = 0; r < 8; ++r) {
    int row = r + half * 8;
    float h = c[r] + bdv;
    hs[row * 65 + w * 16 + l16] = h > 0.f ? h : 0.f;
  }
  __syncthreads();

  // ---- GEMM2: out += Hs @ Wu + bu  (each wave: 32 N-tiles of 16) ----
  for (int nt = w; nt < Hd / 16; nt += 4) {
    v8f d = {};
#pragma unroll
    for (int k0 = 0; k0 < BN; k0 += 4) {
      v2f a;
      a.x = hs[l16 * 65 + k0 + koffA];
      a.y = hs[l16 * 65 + k0 + koffA + 1];
      v2f bf;  // B[k][n] = Wu[k*H + nt*16 + n]
      bf.x = Wu[(size_t)(k0 + koffA) * Hd + nt * 16 + l16];
      bf.y = Wu[(size_t)(k0 + koffA + 1) * Hd + nt * 16 + l16];
      d = __builtin_amdgcn_wmma_f32_16x16x4_f32(false, a, false, bf,
                                                (short)0, d, false, false);
    }
    const float buv = bu[nt * 16 + l16];
#pragma unroll
    for (int r = 0; r < 8; ++r) {
      int row = r + half * 8;
      size_t o = (size_t)gtok[row] * Hd + nt * 16 + l16;
      out[o] += d[r] + buv;  // out already holds x; rows disjoint per tile
    }
  }
}

// ---------------------------------------------------------------------------
extern "C" void kernel_launch(void* const* d_in, const int* in_sizes, int n_in,
                              void* d_out, int out_size, void* d_ws, size_t ws_size,
                              hipStream_t stream) {
  const float* x  = (const float*)d_in[0];
  const float* Wd = (const float*)d_in[1];
  const float* bd = (const float*)d_in[2];
  const float* Wu = (const float*)d_in[3];
  const float* bu = (const float*)d_in[4];
  const float* Wg = (const float*)d_in[5];
  const float* bg = (const float*)d_in[6];
  float* out = (float*)d_out;

  float* logits = (float*)d_ws;                                   // B*T floats
  int* selected = (int*)((char*)d_ws + (size_t)Bb * Tt * sizeof(float));

  // 1 wave per token; 8 waves per 256-thread block
  const int n_tok = Bb * Tt;
  copy_logits_kernel<<<n_tok / 8, 256, 0, stream>>>(x, Wg, bg, out, logits);

  topk_kernel<<<Bb, 1024, 0, stream>>>(logits, selected);

  adapter_kernel<<<(Bb * TOPK) / 16, 128, 0, stream>>>(x, Wd, bd, Wu, bu,
                                                       selected, out);
}